// AlternativeMVN_89807766159362
// MI455X (gfx1250) — compile-verified
//
#include <hip/hip_runtime.h>
#include <math.h>

typedef __attribute__((ext_vector_type(2))) float v2f;
typedef __attribute__((ext_vector_type(8))) float v8f;

#define N_ROWS  4096
#define IN_DIM  2048
#define COVD    4096
#define K4STEPS (IN_DIM / 4)   // 512 WMMA K-steps

// ---------------------------------------------------------------------------
// Kernel A: fold cov_kernel (2048x4096, row-major) against the two rho halves.
//   pa[k] = sum_j cov_k[k,j]*rho[j] ; pb[k] = sum_j cov_k[k,j]*rho[4096+j]
// Virtual row 2048 = cov_bias -> consts[0..1] = cov_bias . {wa, wb}
// ---------------------------------------------------------------------------
__global__ void __launch_bounds__(256)
fold_kernel(const float* __restrict__ cov_k,
            const float* __restrict__ cov_bias,
            const float* __restrict__ rho,
            float* __restrict__ pa, float* __restrict__ pb,
            float* __restrict__ consts) {
  const int wid  = threadIdx.x >> 5;
  const int lane = threadIdx.x & 31;
  const int r = blockIdx.x * 8 + wid;
  if (r > IN_DIM) return;                               // uniform per wave
  const float* rowp = (r < IN_DIM) ? (cov_k + (size_t)r * COVD) : cov_bias;
  const float* wa = rho;
  const float* wb = rho + COVD;
  float aa = 0.f, ab = 0.f;
  for (int j = lane; j < COVD; j += 32) {
    float cv = rowp[j];
    aa = fmaf(cv, wa[j], aa);
    ab = fmaf(cv, wb[j], ab);
  }
  for (int off = 16; off > 0; off >>= 1) {
    aa += __shfl_down(aa, off, 32);
    ab += __shfl_down(ab, off, 32);
  }
  if (lane == 0) {
    if (r < IN_DIM) { pa[r] = aa; pb[r] = ab; }
    else            { consts[0] = aa; consts[1] = ab; }
  }
}

// ---------------------------------------------------------------------------
// Kernel A2: pre-pack the (2048 x 16) B operand into WMMA-fragment order.
// B fragment (4x16 f32): lane l, reg r -> B[K = r + 2*(l>>4), N = l&15]
// For K-step k4: bpack[k4*64 + l*2 + r] = col_{l&15}[4*k4 + r + 2*(l>>4)]
// Columns: 0=mu_kernel, 1=var_kernel, 2=pa, 3=pb, 4..15 = 0.
// One divergent pass over 32K elements; inner GEMM loop becomes branch-free.
// ---------------------------------------------------------------------------
__global__ void __launch_bounds__(256)
pack_b_kernel(const float* __restrict__ mu_k, const float* __restrict__ var_k,
              const float* __restrict__ pa, const float* __restrict__ pb,
              float* __restrict__ bpack) {
  const int tid  = blockIdx.x * 256 + threadIdx.x;   // 16384 threads
  const int k4   = tid >> 5;
  const int lane = tid & 31;
  const int n    = lane & 15;
  const int kk   = k4 * 4 + 2 * (lane >> 4);

  const float* colp = nullptr;
  if      (n == 0) colp = mu_k;
  else if (n == 1) colp = var_k;
  else if (n == 2) colp = pa;
  else if (n == 3) colp = pb;

  v2f b;
  b.x = colp ? colp[kk]     : 0.f;
  b.y = colp ? colp[kk + 1] : 0.f;
  *(v2f*)(bpack + (size_t)k4 * 64 + lane * 2) = b;
}

// ---------------------------------------------------------------------------
// Kernel B: fused head GEMM via V_WMMA_F32_16X16X4_F32.
// (4096 x 2048) @ (2048 x 16), fp32 end-to-end; one wave per 16-row tile.
// A fragment (16x4 f32): lane l, reg r -> A[M=l&15, K=r+2*(l>>4)]
// C/D (16x16 f32):       lane l, reg r -> D[M=r+8*(l>>4), N=l&15]
// Inner loop: 2 coalesced b64 loads + 1 WMMA, no divergence.
// ---------------------------------------------------------------------------
__global__ void __launch_bounds__(256)
head_gemm_wmma(const float* __restrict__ x,
               const float* __restrict__ bpack,
               const float* __restrict__ mu_b, const float* __restrict__ var_b,
               const float* __restrict__ consts,
               float* __restrict__ mu_out,
               float* __restrict__ uvec, float* __restrict__ vvec,
               float* __restrict__ dvec) {
  const int wid   = threadIdx.x >> 5;
  const int lane  = threadIdx.x & 31;
  const int tileM = (blockIdx.x * 8 + wid) * 16;
  const int n     = lane & 15;
  const int hi    = lane >> 4;          // 0 or 1
  const int koff  = hi * 2;

  const float* xrow = x + (size_t)(tileM + n) * IN_DIM + koff;
  const float* bp   = bpack + lane * 2;

  v8f c = {};
#pragma unroll 4
  for (int k4 = 0; k4 < K4STEPS; ++k4) {
    v2f a = *(const v2f*)(xrow + k4 * 4);
    v2f b = *(const v2f*)(bp   + k4 * 64);
    c = __builtin_amdgcn_wmma_f32_16x16x4_f32(false, a, false, b,
                                              (short)0, c, false, false);
  }

  const float ca = consts[0], cb = consts[1];
  const float mb = mu_b[0],  vb = var_b[0];
#pragma unroll
  for (int r = 0; r < 8; ++r) {
    const int row = tileM + r + 8 * hi;
    const float val = c[r];
    if      (n == 0) mu_out[row] = val + mb;
    else if (n == 1) dvec[row]   = log1pf(expf(val + vb)) + 1e-8f;
    else if (n == 2) uvec[row]   = val + ca;
    else if (n == 3) vvec[row]   = val + cb;
  }
}

// ---------------------------------------------------------------------------
// Kernel C: inclusive prefix sums of v and v^2 over 4096 elements.
// Single block of 1024 threads, 4 elems/thread, LDS Hillis-Steele scan.
// ---------------------------------------------------------------------------
__global__ void __launch_bounds__(1024)
scan_kernel(const float* __restrict__ vvec,
            float* __restrict__ S1, float* __restrict__ S2) {
  __shared__ float s1[2][1024];
  __shared__ float s2[2][1024];
  const int tid = threadIdx.x;

  float l1[4], l2[4];
  float t1 = 0.f, t2 = 0.f;
#pragma unroll
  for (int i = 0; i < 4; ++i) {
    const float v = vvec[tid * 4 + i];
    t1 += v; t2 = fmaf(v, v, t2);
    l1[i] = t1; l2[i] = t2;            // inclusive within thread
  }
  s1[0][tid] = t1; s2[0][tid] = t2;
  __syncthreads();

  int src = 0;
  for (int off = 1; off < 1024; off <<= 1) {
    float a1 = s1[src][tid], a2 = s2[src][tid];
    if (tid >= off) { a1 += s1[src][tid - off]; a2 += s2[src][tid - off]; }
    s1[src ^ 1][tid] = a1; s2[src ^ 1][tid] = a2;
    __syncthreads();
    src ^= 1;
  }
  const float e1 = s1[src][tid] - t1;  // exclusive prefix of thread totals
  const float e2 = s2[src][tid] - t2;
#pragma unroll
  for (int i = 0; i < 4; ++i) {
    S1[tid * 4 + i] = e1 + l1[i];
    S2[tid * 4 + i] = e2 + l2[i];
  }
}

// ---------------------------------------------------------------------------
// Kernel D: closed-form fill of output_cov (4096x4096) + diagonal scatter.
//   m = min(i,j);  mm[i,j] = (m+1)*u_i*u_j + (u_i+u_j)*S1[m] + S2[m]
//   out[i,i] = d[i]
// float4 streaming stores; this 67 MB write dominates total runtime.
// ---------------------------------------------------------------------------
__global__ void __launch_bounds__(256)
cov_fill_kernel(const float* __restrict__ u,
                const float* __restrict__ S1, const float* __restrict__ S2,
                const float* __restrict__ dvec,
                float* __restrict__ out_cov) {
  const int i  = blockIdx.y;
  const int j0 = (blockIdx.x * 256 + threadIdx.x) * 4;
  const float ui  = u[i];
  const float s1i = S1[i], s2i = S2[i];        // used when j >= i (m = i)
  float4 out;
  float* o = (float*)&out;
#pragma unroll
  for (int t = 0; t < 4; ++t) {
    const int j = j0 + t;
    const float uj = u[j];
    float s1, s2, mp1;
    if (j < i) { s1 = S1[j]; s2 = S2[j]; mp1 = (float)(j + 1); }
    else       { s1 = s1i;   s2 = s2i;   mp1 = (float)(i + 1); }
    float val = mp1 * ui * uj + (ui + uj) * s1 + s2;
    if (j == i) val = dvec[i];
    o[t] = val;
  }
  *(float4*)(out_cov + (size_t)i * COVD + j0) = out;
}

// ---------------------------------------------------------------------------
extern "C" void kernel_launch(void* const* d_in, const int* in_sizes, int n_in,
                              void* d_out, int out_size, void* d_ws, size_t ws_size,
                              hipStream_t stream) {
  (void)in_sizes; (void)n_in; (void)out_size; (void)ws_size;
  const float* x        = (const float*)d_in[0];
  const float* mu_k     = (const float*)d_in[1];
  const float* mu_b     = (const float*)d_in[2];
  const float* cov_k    = (const float*)d_in[3];
  const float* cov_bias = (const float*)d_in[4];
  const float* var_k    = (const float*)d_in[5];
  const float* var_b    = (const float*)d_in[6];
  const float* rho      = (const float*)d_in[7];

  float* ws     = (float*)d_ws;
  float* pa     = ws;            // 2048
  float* pb     = ws + 2048;     // 2048
  float* consts = ws + 4096;     // 2
  float* u      = ws + 4104;     // 4096
  float* v      = ws + 8200;     // 4096
  float* dv     = ws + 12296;    // 4096
  float* S1     = ws + 16392;    // 4096
  float* S2     = ws + 20488;    // 4096
  float* bpack  = ws + 24584;    // 32768 (128 KB packed B fragments)

  float* mu_out  = (float*)d_out;            // output_mu: 4096 floats
  float* cov_out = (float*)d_out + N_ROWS;   // output_cov: 4096*4096 floats

  // A: fold cov_kernel with rho halves (+ bias row -> consts)
  fold_kernel<<<257, 256, 0, stream>>>(cov_k, cov_bias, rho, pa, pb, consts);
  // A2: pack B operand into WMMA fragment order (branch-free inner loop)
  pack_b_kernel<<<64, 256, 0, stream>>>(mu_k, var_k, pa, pb, bpack);
  // B: fused head GEMM via WMMA f32 16x16x4 -> mu_out, u, v, d
  head_gemm_wmma<<<32, 256, 0, stream>>>(x, bpack, mu_b, var_b, consts,
                                         mu_out, u, v, dv);
  // C: prefix sums of v, v^2
  scan_kernel<<<1, 1024, 0, stream>>>(v, S1, S2);
  // D: closed-form covariance fill + diagonal
  dim3 gridD(COVD / 1024, N_ROWS);
  cov_fill_kernel<<<gridD, 256, 0, stream>>>(u, S1, S2, dv, cov_out);
}